// GATConvEncoder_55362128445915
// MI455X (gfx1250) — compile-verified
//
#include <hip/hip_runtime.h>
#include <hip/hip_bf16.h>

typedef __attribute__((ext_vector_type(2))) float v2f;
typedef __attribute__((ext_vector_type(8))) float v8f;

#define GN   20000
#define GE   320000
#define GH   4
#define GCH  128
#define GF   512
#define GOUT 32
#define NEG_SLOPE 0.2f
#define BN_EPS 1e-5f

#define BM 128      // block tile rows
#define BN_T 64     // block tile cols
#define BK 32       // K chunk staged in LDS

// ---------------- fp32 WMMA GEMM: C[M,Nc] = A[M,K] @ B[K,Nc] -------------
// 256 threads = 8 waves; waves in 4x2 grid; each wave owns a 32x32 micro-tile
// (four 16x16 fp32 WMMA accumulators). B tile is stored transposed in LDS so
// every fragment load is a contiguous 64-bit DS read.
__global__ __launch_bounds__(256) void gemm_wmma_f32(
    const float* __restrict__ A, const float* __restrict__ B,
    float* __restrict__ C, int M, int K, int Nc)
{
    __shared__ float sA [BM][BK + 1];    // 128x33 fl = 16.9 KB
    __shared__ float sBt[BN_T][BK + 1];  //  64x33 fl =  8.4 KB (transposed)

    const int tid  = threadIdx.x;
    const int lane = tid & 31;
    const int wave = tid >> 5;       // 0..7
    const int wm   = wave & 3;       // 4 row groups of 32
    const int wn   = wave >> 2;      // 2 col groups of 32
    const int brow = blockIdx.x * BM;
    const int bcol = blockIdx.y * BN_T;

    v8f acc00 = {}, acc01 = {}, acc10 = {}, acc11 = {};

    const int khalf = (lane >> 4) ? 2 : 0;        // lanes 16-31 hold K=2,3
    const int arow0 = wm * 32 + (lane & 15);
    const int bcol0 = wn * 32 + (lane & 15);

    for (int k0 = 0; k0 < K; k0 += BK) {
        // cooperative loads into LDS (global reads coalesced)
        #pragma unroll
        for (int i = tid; i < BM * BK; i += 256) {
            int r = i / BK, c = i % BK;
            int gr = brow + r;
            sA[r][c] = (gr < M) ? A[(size_t)gr * K + k0 + c] : 0.0f;
        }
        #pragma unroll
        for (int i = tid; i < BK * BN_T; i += 256) {
            int r = i / BN_T, c = i % BN_T;
            int gc = bcol + c;
            sBt[c][r] = (gc < Nc) ? B[(size_t)(k0 + r) * Nc + gc] : 0.0f;
        }
        __syncthreads();

        #pragma unroll
        for (int kk = 0; kk < BK; kk += 4) {
            const int ak = kk + khalf;
            v2f a0, a1, b0, b1;
            a0.x = sA[arow0][ak];           a0.y = sA[arow0][ak + 1];
            a1.x = sA[arow0 + 16][ak];      a1.y = sA[arow0 + 16][ak + 1];
            b0.x = sBt[bcol0][ak];          b0.y = sBt[bcol0][ak + 1];
            b1.x = sBt[bcol0 + 16][ak];     b1.y = sBt[bcol0 + 16][ak + 1];
            acc00 = __builtin_amdgcn_wmma_f32_16x16x4_f32(
                false, a0, false, b0, (short)0, acc00, false, false);
            acc01 = __builtin_amdgcn_wmma_f32_16x16x4_f32(
                false, a0, false, b1, (short)0, acc01, false, false);
            acc10 = __builtin_amdgcn_wmma_f32_16x16x4_f32(
                false, a1, false, b0, (short)0, acc10, false, false);
            acc11 = __builtin_amdgcn_wmma_f32_16x16x4_f32(
                false, a1, false, b1, (short)0, acc11, false, false);
        }
        __syncthreads();
    }

    // C/D layout: VGPR i -> lanes 0-15: (M=i, N=lane) ; lanes 16-31: (M=i+8, N=lane-16)
    const int crow = brow + wm * 32 + ((lane >> 4) ? 8 : 0);
    const int ccol = bcol + wn * 32 + (lane & 15);
    #pragma unroll
    for (int i = 0; i < 8; ++i) {
        int r0 = crow + i;
        int r1 = r0 + 16;
        if (r0 < M) {
            if (ccol < Nc)      C[(size_t)r0 * Nc + ccol]      = acc00[i];
            if (ccol + 16 < Nc) C[(size_t)r0 * Nc + ccol + 16] = acc01[i];
        }
        if (r1 < M) {
            if (ccol < Nc)      C[(size_t)r1 * Nc + ccol]      = acc10[i];
            if (ccol + 16 < Nc) C[(size_t)r1 * Nc + ccol + 16] = acc11[i];
        }
    }
}

// ---------------- per-(node,head) attention logits ----------------------
__global__ void alpha_kernel(const float* __restrict__ h,
                             const float* __restrict__ a_src,
                             const float* __restrict__ a_dst,
                             float* __restrict__ als, float* __restrict__ ald,
                             int Nn, int heads, int ch)
{
    int t = blockIdx.x * blockDim.x + threadIdx.x;
    if (t >= Nn * heads) return;
    int n = t / heads, hd = t % heads;
    const float* row = h + (size_t)n * heads * ch + (size_t)hd * ch;
    const float* as  = a_src + (size_t)hd * ch;
    const float* ad  = a_dst + (size_t)hd * ch;
    float s = 0.f, d = 0.f;
    for (int c = 0; c < ch; ++c) {
        float v = row[c];
        s += v * as[c];
        d += v * ad[c];
    }
    als[t] = s;
    ald[t] = d;
}

// ---------------- fills --------------------------------------------------
__global__ void fill_u32(unsigned* p, unsigned v, size_t n)
{
    size_t t = (size_t)blockIdx.x * blockDim.x + threadIdx.x;
    if (t < n) p[t] = v;
}
__global__ void fill_f32(float* p, float v, size_t n)
{
    size_t t = (size_t)blockIdx.x * blockDim.x + threadIdx.x;
    if (t < n) p[t] = v;
}

// order-preserving float <-> uint mapping for atomic segment-max
__device__ __forceinline__ unsigned f2ord(float f)
{
    unsigned u = __float_as_uint(f);
    return (u & 0x80000000u) ? ~u : (u | 0x80000000u);
}
__device__ __forceinline__ float ord2f(unsigned u)
{
    return __uint_as_float((u & 0x80000000u) ? (u & 0x7FFFFFFFu) : ~u);
}

// ---------------- edge pass 1: leaky-relu logits + segment max ----------
__global__ void edge_max_kernel(const int* __restrict__ ei, int E, int Nn, int heads,
                                const float* __restrict__ als,
                                const float* __restrict__ ald,
                                float* __restrict__ ebuf, unsigned* __restrict__ mmax)
{
    int t = blockIdx.x * blockDim.x + threadIdx.x;
    int tot = (E + Nn) * heads;
    if (t >= tot) return;
    int e = t / heads, hd = t % heads;
    int src = (e < E) ? ei[e]     : (e - E);
    int dst = (e < E) ? ei[E + e] : (e - E);
    float v = als[src * heads + hd] + ald[dst * heads + hd];
    v = (v > 0.f) ? v : v * NEG_SLOPE;
    ebuf[t] = v;
    atomicMax(&mmax[dst * heads + hd], f2ord(v));
}

// ---------------- edge pass 2: exp + segment sum ------------------------
__global__ void edge_exp_kernel(const int* __restrict__ ei, int E, int Nn, int heads,
                                const unsigned* __restrict__ mmax,
                                float* __restrict__ ebuf, float* __restrict__ ssum)
{
    int t = blockIdx.x * blockDim.x + threadIdx.x;
    int tot = (E + Nn) * heads;
    if (t >= tot) return;
    int e = t / heads, hd = t % heads;
    int dst = (e < E) ? ei[E + e] : (e - E);
    float m  = ord2f(mmax[dst * heads + hd]);
    float ex = __expf(ebuf[t] - m);
    ebuf[t]  = ex;
    atomicAdd(&ssum[dst * heads + hd], ex);
}

// ---------------- edge pass 3: weighted scatter-add (wave per edge) -----
__global__ __launch_bounds__(256) void edge_aggregate_kernel(
    const int* __restrict__ ei, int E, int Nn, int heads, int ch,
    const float* __restrict__ hfeat, const float* __restrict__ ebuf,
    const float* __restrict__ ssum, float* __restrict__ out)
{
    int gwave = (int)(((size_t)blockIdx.x * blockDim.x + threadIdx.x) >> 5);
    int lane  = threadIdx.x & 31;
    int E2 = E + Nn;
    if (gwave >= E2) return;
    int src = (gwave < E) ? ei[gwave]     : (gwave - E);
    int dst = (gwave < E) ? ei[E + gwave] : (gwave - E);
    int F = heads * ch;
    for (int c = lane; c < F; c += 32) {
        int hd = c / ch;
        float alpha = ebuf[gwave * heads + hd] / ssum[dst * heads + hd];
        atomicAdd(&out[(size_t)dst * F + c], alpha * hfeat[(size_t)src * F + c]);
    }
}

// ---------------- bias + BN(eval) + ReLU (+ residual ops) ---------------
__global__ void post_kernel(float* __restrict__ h, const float* __restrict__ bias,
                            const float* __restrict__ g, const float* __restrict__ be,
                            const float* __restrict__ resid_in,
                            float* __restrict__ resid_out, int Nn, int F)
{
    size_t t = (size_t)blockIdx.x * blockDim.x + threadIdx.x;
    if (t >= (size_t)Nn * F) return;
    int c = (int)(t % F);
    float v = h[t] + bias[c];
    v = v * (g[c] * rsqrtf(1.0f + BN_EPS)) + be[c];
    v = fmaxf(v, 0.0f);
    if (resid_in) v += resid_in[t];
    h[t] = v;
    if (resid_out) resid_out[t] = v;
}

__global__ void final_bias_kernel(const float* __restrict__ agg,
                                  const float* __restrict__ bo,
                                  float* __restrict__ z, int Nn, int F)
{
    size_t t = (size_t)blockIdx.x * blockDim.x + threadIdx.x;
    if (t >= (size_t)Nn * F) return;
    z[t] = agg[t] + bo[t % F];
}

// ------------------------------------------------------------------------
extern "C" void kernel_launch(void* const* d_in, const int* in_sizes, int n_in,
                              void* d_out, int out_size, void* d_ws, size_t ws_size,
                              hipStream_t stream)
{
    const float* x   = (const float*)d_in[0];
    const int*   ei  = (const int*)  d_in[1];
    const float* W1  = (const float*)d_in[2];
    const float* as1 = (const float*)d_in[3];
    const float* ad1 = (const float*)d_in[4];
    const float* b1  = (const float*)d_in[5];
    const float* g1  = (const float*)d_in[6];
    const float* be1 = (const float*)d_in[7];
    const float* W2  = (const float*)d_in[8];
    const float* as2 = (const float*)d_in[9];
    const float* ad2 = (const float*)d_in[10];
    const float* b2  = (const float*)d_in[11];
    const float* g2  = (const float*)d_in[12];
    const float* be2 = (const float*)d_in[13];
    const float* Wo  = (const float*)d_in[14];
    const float* aso = (const float*)d_in[15];
    const float* ado = (const float*)d_in[16];
    const float* bo  = (const float*)d_in[17];
    float* z = (float*)d_out;

    const int Nn = GN, E = GE, E2 = GE + GN;

    // workspace carve-up (fp32 words)
    float*    P    = (float*)d_ws;               // [N,512] GEMM out / agg src
    float*    Q    = P + (size_t)Nn * GF;        // [N,512] agg dst / next input
    float*    R    = Q + (size_t)Nn * GF;        // [N,512] residual / L3 agg
    float*    als  = R + (size_t)Nn * GF;        // [N,4]
    float*    ald  = als + (size_t)Nn * GH;      // [N,4]
    unsigned* mmax = (unsigned*)(ald + (size_t)Nn * GH);  // [N,4]
    float*    ssum = (float*)mmax + (size_t)Nn * GH;      // [N,4]
    float*    ebuf = ssum + (size_t)Nn * GH;     // [E2,4]

    const dim3 blk(256);
    const int gRow = (Nn + BM - 1) / BM;                  // 157
    #define CEIL(a) ((unsigned)(((a) + 255) / 256))

    // =================== Layer 1: GATConv(512->4x128) ===================
    gemm_wmma_f32<<<dim3(gRow, GF / BN_T), blk, 0, stream>>>(x, W1, P, Nn, GF, GF);
    alpha_kernel<<<CEIL(Nn * GH), blk, 0, stream>>>(P, as1, ad1, als, ald, Nn, GH, GCH);
    fill_u32<<<CEIL(Nn * GH), blk, 0, stream>>>(mmax, 0u, (size_t)Nn * GH);
    fill_f32<<<CEIL(Nn * GH), blk, 0, stream>>>(ssum, 0.f, (size_t)Nn * GH);
    fill_f32<<<CEIL((size_t)Nn * GF), blk, 0, stream>>>(Q, 0.f, (size_t)Nn * GF);
    edge_max_kernel<<<CEIL(E2 * GH), blk, 0, stream>>>(ei, E, Nn, GH, als, ald, ebuf, mmax);
    edge_exp_kernel<<<CEIL(E2 * GH), blk, 0, stream>>>(ei, E, Nn, GH, mmax, ebuf, ssum);
    edge_aggregate_kernel<<<CEIL((size_t)E2 * 32), blk, 0, stream>>>(
        ei, E, Nn, GH, GCH, P, ebuf, ssum, Q);
    post_kernel<<<CEIL((size_t)Nn * GF), blk, 0, stream>>>(Q, b1, g1, be1, nullptr, R, Nn, GF);

    // =================== Layer 2: GATConv(512->4x128) + residual ========
    gemm_wmma_f32<<<dim3(gRow, GF / BN_T), blk, 0, stream>>>(Q, W2, P, Nn, GF, GF);
    alpha_kernel<<<CEIL(Nn * GH), blk, 0, stream>>>(P, as2, ad2, als, ald, Nn, GH, GCH);
    fill_u32<<<CEIL(Nn * GH), blk, 0, stream>>>(mmax, 0u, (size_t)Nn * GH);
    fill_f32<<<CEIL(Nn * GH), blk, 0, stream>>>(ssum, 0.f, (size_t)Nn * GH);
    fill_f32<<<CEIL((size_t)Nn * GF), blk, 0, stream>>>(Q, 0.f, (size_t)Nn * GF);
    edge_max_kernel<<<CEIL(E2 * GH), blk, 0, stream>>>(ei, E, Nn, GH, als, ald, ebuf, mmax);
    edge_exp_kernel<<<CEIL(E2 * GH), blk, 0, stream>>>(ei, E, Nn, GH, mmax, ebuf, ssum);
    edge_aggregate_kernel<<<CEIL((size_t)E2 * 32), blk, 0, stream>>>(
        ei, E, Nn, GH, GCH, P, ebuf, ssum, Q);
    post_kernel<<<CEIL((size_t)Nn * GF), blk, 0, stream>>>(Q, b2, g2, be2, R, nullptr, Nn, GF);

    // =================== Output head: GATConv(512->1x32, mean) ==========
    gemm_wmma_f32<<<dim3(gRow, 1), blk, 0, stream>>>(Q, Wo, P, Nn, GF, GOUT);
    alpha_kernel<<<CEIL(Nn), blk, 0, stream>>>(P, aso, ado, als, ald, Nn, 1, GOUT);
    fill_u32<<<CEIL(Nn), blk, 0, stream>>>(mmax, 0u, (size_t)Nn);
    fill_f32<<<CEIL(Nn), blk, 0, stream>>>(ssum, 0.f, (size_t)Nn);
    fill_f32<<<CEIL((size_t)Nn * GOUT), blk, 0, stream>>>(R, 0.f, (size_t)Nn * GOUT);
    edge_max_kernel<<<CEIL(E2), blk, 0, stream>>>(ei, E, Nn, 1, als, ald, ebuf, mmax);
    edge_exp_kernel<<<CEIL(E2), blk, 0, stream>>>(ei, E, Nn, 1, mmax, ebuf, ssum);
    edge_aggregate_kernel<<<CEIL((size_t)E2 * 32), blk, 0, stream>>>(
        ei, E, Nn, 1, GOUT, P, ebuf, ssum, R);
    final_bias_kernel<<<CEIL((size_t)Nn * GOUT), blk, 0, stream>>>(R, bo, z, Nn, GOUT);
    #undef CEIL
}